// NaturalCubic_69724499083960
// MI455X (gfx1250) — compile-verified
//
#include <hip/hip_runtime.h>
#include <stdint.h>

// NaturalCubic spline evaluation, refactored:
//   out[p] = ((A[j]*e + B[j])*e + C[j])*e + D[j],  j = #{sorted knots < e}
// Coefficient set selected by (p % 3) within each batch; 66 params per batch
// staged into LDS via the CDNA5 Tensor Data Mover (tensor_load_to_lds).

#define NK 10
#define NCH 3
#define PARAMS_PER_B 66            // 2*NCH*NK + 2*NCH
#define BATCH_FLOATS (NCH * 448 * 448)  // 602112
#define TPB 256
#define CHUNK 12                   // floats per thread per iteration (4 x float3-phase)
#define ITERS 2                    // iterations per block -> 98 blocks per batch

typedef float    f32x4 __attribute__((ext_vector_type(4)));
typedef uint32_t u32x4 __attribute__((ext_vector_type(4)));
typedef int      i32x8 __attribute__((ext_vector_type(8)));
typedef int      i32x4 __attribute__((ext_vector_type(4)));

__device__ __forceinline__ float uniform_f(float x) {
#if defined(__HIP_DEVICE_COMPILE__)
  return __int_as_float(__builtin_amdgcn_readfirstlane(__float_as_int(x)));
#else
  return x;
#endif
}

__device__ __forceinline__ void wait_tensorcnt0() {
#if defined(__HIP_DEVICE_COMPILE__)
#if __has_builtin(__builtin_amdgcn_s_wait_tensorcnt)
  __builtin_amdgcn_s_wait_tensorcnt(0);
#else
  asm volatile("s_wait_tensorcnt 0x0" ::: "memory");
#endif
#endif
}

__global__ __launch_bounds__(TPB) void NaturalCubic_kernel(
    const float* __restrict__ raw, const float* __restrict__ params,
    float* __restrict__ out) {
  __shared__ __align__(16) float s_params[72];      // 66 used
  __shared__ float s_sorted[NCH][NK];               // sorted knot positions
  __shared__ f32x4 s_table[NCH][NK + 1];            // {A,B,C,D} prefix tables

  const int b = blockIdx.y;
  const float* pb = params + (size_t)b * PARAMS_PER_B;

  // ---- Stage 66 batch params into LDS via the Tensor Data Mover ----
#if defined(__HIP_DEVICE_COMPILE__) && __has_builtin(__builtin_amdgcn_tensor_load_to_lds)
  if (threadIdx.x == 0) {
    uint64_t ga = (uint64_t)(uintptr_t)pb;
    uint32_t la = (uint32_t)(uintptr_t)&s_params[0];
    // D# group0: count=1 | lds_addr | global_addr[56:0] | type=2
    u32x4 g0;
    g0[0] = 1u;
    g0[1] = la;
    g0[2] = (uint32_t)ga;
    g0[3] = ((uint32_t)(ga >> 32) & 0x01FFFFFFu) | 0x80000000u;
    // D# group1: data_size=4B; tensor_dim0=66; tensor_dim1=1; tile_dim0=66 (1D tile)
    i32x8 g1;
    g1[0] = 0x20000;            // data_size=2 (4 bytes), no multicast/pad/iterate
    g1[1] = (int)(66u << 16);   // tensor_dim0[15:0]
    g1[2] = (int)(1u << 16);    // tensor_dim0[31:16]=0 | tensor_dim1[15:0]=1
    g1[3] = (int)(66u << 16);   // tensor_dim1[31:16]=0 | tile_dim0=66
    g1[4] = 0;                  // tile_dim1=0 (unused), tile_dim2=0
    g1[5] = 66;                 // tensor_dim0_stride[31:0]
    g1[6] = 0;
    g1[7] = 0;
    i32x4 g2 = {0, 0, 0, 0};
    i32x4 g3 = {0, 0, 0, 0};
#if __has_include(<hip/amd_detail/amd_gfx1250_TDM.h>)
    i32x8 g4 = {0, 0, 0, 0, 0, 0, 0, 0};  // amdgpu-toolchain 6-arg form
    __builtin_amdgcn_tensor_load_to_lds(g0, g1, g2, g3, g4, 0);
#else
    __builtin_amdgcn_tensor_load_to_lds(g0, g1, g2, g3, 0);  // ROCm 7.2 5-arg form
#endif
    wait_tensorcnt0();
  }
#else
  if (threadIdx.x < PARAMS_PER_B) s_params[threadIdx.x] = pb[threadIdx.x];
#endif
  __syncthreads();

  // ---- Build sorted-knot prefix tables (3 threads, one per channel phase) ----
  if (threadIdx.x < NCH) {
    const int c = threadIdx.x;
    float xs[NK], al[NK];
    for (int k = 0; k < NK; ++k) {
      xs[k] = s_params[k * NCH + c];
      al[k] = s_params[NCH * NK + k * NCH + c];
    }
    const float bias  = s_params[2 * NCH * NK + c];
    const float slope = s_params[2 * NCH * NK + NCH + c];
    // insertion sort (knots,alphas) ascending by knot position
    for (int i = 1; i < NK; ++i) {
      float x = xs[i], a = al[i];
      int j = i - 1;
      while (j >= 0 && xs[j] > x) { xs[j + 1] = xs[j]; al[j + 1] = al[j]; --j; }
      xs[j + 1] = x; al[j + 1] = a;
    }
    // suffix sums for cubic/quadratic coefs, prefix sums for linear/const coefs
    float A[NK + 1], Bv[NK + 1];
    A[NK] = 0.f; Bv[NK] = 0.f;
    for (int j = NK - 1; j >= 0; --j) {
      A[j]  = A[j + 1] - al[j] * (1.0f / 6.0f);
      Bv[j] = Bv[j + 1] + 0.5f * al[j] * xs[j];
    }
    float Cv = slope, Dv = bias;
    for (int j = 0; j <= NK; ++j) {
      f32x4 t; t[0] = A[j]; t[1] = Bv[j]; t[2] = Cv; t[3] = Dv;
      s_table[c][j] = t;
      if (j < NK) {
        Cv += 0.5f * al[j] * xs[j] * xs[j];
        Dv -= al[j] * xs[j] * xs[j] * xs[j] * (1.0f / 6.0f);
      }
    }
    for (int k = 0; k < NK; ++k) s_sorted[c][k] = xs[k];
  }
  __syncthreads();

  // Pin the 30 sorted knot positions into SGPRs (uniform across the block).
  float sx[NCH][NK];
#pragma unroll
  for (int c = 0; c < NCH; ++c)
#pragma unroll
    for (int k = 0; k < NK; ++k) sx[c][k] = uniform_f(s_sorted[c][k]);

  // ---- Streaming evaluation: 12 floats/thread so the %3 phase is static ----
  const size_t batchBase = (size_t)b * BATCH_FLOATS;
  for (int it = 0; it < ITERS; ++it) {
    size_t base = batchBase +
        ((size_t)(blockIdx.x * ITERS + it) * TPB + threadIdx.x) * CHUNK;
    const f32x4* src = (const f32x4*)(raw + base);
    f32x4 v0 = __builtin_nontemporal_load(src + 0);
    f32x4 v1 = __builtin_nontemporal_load(src + 1);
    f32x4 v2 = __builtin_nontemporal_load(src + 2);

    float e[CHUNK] __attribute__((aligned(16)));
    *(f32x4*)&e[0] = v0; *(f32x4*)&e[4] = v1; *(f32x4*)&e[8] = v2;
    float r[CHUNK] __attribute__((aligned(16)));

#pragma unroll
    for (int j = 0; j < CHUNK; ++j) {
      const int c = j % NCH;  // compile-time channel phase
      const float xe = e[j];
      int cnt = 0;
#pragma unroll
      for (int k = 0; k < NK; ++k) cnt += (xe > sx[c][k]) ? 1 : 0;
      f32x4 t = s_table[c][cnt];  // ds_load_b128
      r[j] = fmaf(fmaf(fmaf(t[0], xe, t[1]), xe, t[2]), xe, t[3]);
    }

    f32x4* dst = (f32x4*)(out + base);
    __builtin_nontemporal_store(*(f32x4*)&r[0], dst + 0);
    __builtin_nontemporal_store(*(f32x4*)&r[4], dst + 1);
    __builtin_nontemporal_store(*(f32x4*)&r[8], dst + 2);
  }
}

extern "C" void kernel_launch(void* const* d_in, const int* in_sizes, int n_in,
                              void* d_out, int out_size, void* d_ws, size_t ws_size,
                              hipStream_t stream) {
  const float* raw    = (const float*)d_in[0];
  const float* params = (const float*)d_in[1];
  float* out          = (float*)d_out;

  const int B = in_sizes[1] / PARAMS_PER_B;                 // 16
  const int blocksX = BATCH_FLOATS / (TPB * CHUNK * ITERS); // 602112/6144 = 98
  dim3 grid(blocksX, B);
  NaturalCubic_kernel<<<grid, TPB, 0, stream>>>(raw, params, out);
}